// ApproximatorLossFn_76673756168427
// MI455X (gfx1250) — compile-verified
//
#include <hip/hip_runtime.h>

typedef __attribute__((ext_vector_type(2))) float v2f;
typedef __attribute__((ext_vector_type(8))) float v8f;

#define PADV     (-10000.0f)
#define SENT     (-100.0f)
#define EPSV     0.0025f
#define INV_EPS  400.0f
#define LOG2E    1.4426950408889634f
#define LN2      0.6931471805599453f
#define NITER    30
#define BATCH    64
#define TLEN     512
#define WLEN     510

// Pure-VALU half-wave swap: lane i <-> lane i^16 (wave32), via v_permlanex16_b32
// with identity nibble selects (avoids ds_bpermute + s_wait_dscnt in the hot loop).
__device__ __forceinline__ float xor16_f32(float v) {
  int i = __float_as_int(v);
  int r = __builtin_amdgcn_permlanex16(i, i, 0x76543210, 0xfedcba98, false, false);
  return __int_as_float(r);
}

// One softmin pass over the full 512 (padded) problem.
//   out[i] = 0.5*tgt[i]^2 - EPS*ln2*( m2 + log2( sum_j 2^(q'_j + tgt_i*src_j*INV_EPS*LOG2E - m2) ) )
// with q'_j = (wl[j] + (pot[j] - 0.5*src[j]^2)*INV_EPS)*LOG2E  (j<510), else -3e38 (dead).
// V_WMMA_F32_16X16X4_F32 computes u = q + x*y*scale directly: A holds src (M=j rows,
// K=0 col), B holds tgt*scale (K=0 row, N=i cols), and C carries q_j — whose D-layout
// (VGPR k <-> M=k / M=k+8 per lane-half) matches q[j0 + half*8 + k] exactly.
__device__ __forceinline__ void softmin_pass(const float* __restrict__ tgt,
                                             const float* __restrict__ src,
                                             const float* __restrict__ wl,
                                             const float* __restrict__ pot,
                                             float* __restrict__ out,
                                             float* __restrict__ qv,
                                             int tid) {
  // phase 1: build q' (all 256 threads)
  for (int j = tid; j < TLEN; j += 256) {
    float q;
    if (j < WLEN) {
      float s = src[j];
      q = (wl[j] + (pot[j] - 0.5f * s * s) * INV_EPS) * LOG2E;
    } else {
      q = -3.0e38f;  // dead padding column: exp2 -> 0
    }
    qv[j] = q;
  }
  __syncthreads();

  const int lane = tid & 31;
  const int wid  = tid >> 5;
  const int half = lane >> 4;   // 0: j-rows 0..7 of tile, 1: j-rows 8..15
  const int lc   = lane & 15;   // column (i) within tile

  // phase 2: each wave handles 16 output columns per tile, 4 tiles total
  for (int ct = wid; ct < 32; ct += 8) {
    const int i0 = ct << 4;
    // B matrix (4x16): only K=0 row nonzero = tgt * scale  (v0 lanes 0..15)
    const float b0 = (half == 0) ? tgt[i0 + lc] * (INV_EPS * LOG2E) : 0.0f;
    const v2f Bm = {b0, 0.0f};
    float mrun = -3.0e38f;
    float srun = 0.0f;
    for (int jt = 0; jt < 32; ++jt) {
      const int j0 = jt << 4;
      // A matrix (16x4): only K=0 column nonzero = src  (v0 lanes 0..15 hold M=0..15)
      const float a0 = (half == 0) ? src[j0 + lc] : 0.0f;
      const v2f Am = {a0, 0.0f};
      // C matrix: q_j, matching D layout per lane-half (32B-aligned v8f LDS load)
      const v8f qc = *(const v8f*)(qv + j0 + (half << 3));
      // u[M=jj][N=ic] = src[j0+jj]*tgt[i0+ic]*INV_EPS*LOG2E + q[j0+jj]
      v8f u = __builtin_amdgcn_wmma_f32_16x16x4_f32(false, Am, false, Bm,
                                                    (short)0, qc, false, false);
      float tmax = fmaxf(fmaxf(fmaxf(u[0], u[1]), fmaxf(u[2], u[3])),
                         fmaxf(fmaxf(u[4], u[5]), fmaxf(u[6], u[7])));
      tmax = fmaxf(tmax, xor16_f32(tmax));           // merge lane halves (VALU)
      const float mnew = fmaxf(mrun, tmax);
      const float sc = __builtin_amdgcn_exp2f(mrun - mnew);  // online rescale
      float e = (__builtin_amdgcn_exp2f(u[0] - mnew) + __builtin_amdgcn_exp2f(u[1] - mnew))
              + (__builtin_amdgcn_exp2f(u[2] - mnew) + __builtin_amdgcn_exp2f(u[3] - mnew))
              + (__builtin_amdgcn_exp2f(u[4] - mnew) + __builtin_amdgcn_exp2f(u[5] - mnew))
              + (__builtin_amdgcn_exp2f(u[6] - mnew) + __builtin_amdgcn_exp2f(u[7] - mnew));
      srun = fmaf(srun, sc, e);
      mrun = mnew;
    }
    srun += xor16_f32(srun);                         // both halves -> full sum
    const float ti = tgt[i0 + lc];
    const float lse2 = mrun + __builtin_amdgcn_logf(srun);   // v_log_f32 = log2
    const float res = 0.5f * ti * ti - (EPSV * LN2) * lse2;
    if (half == 0) out[i0 + lc] = res;
  }
  __syncthreads();
}

__global__ __launch_bounds__(256) void sinkhorn_kernel(
    const float* __restrict__ y_pred, const float* __restrict__ y_true,
    float* __restrict__ ws) {
  __shared__ __align__(32) float xs[TLEN];
  __shared__ __align__(32) float ys[TLEN];
  __shared__ __align__(32) float wl[TLEN];
  __shared__ __align__(32) float qv[TLEN];
  __shared__ __align__(32) float fa[TLEN];
  __shared__ __align__(32) float ga[TLEN];
  __shared__ float red[256];
  __shared__ int  redi[512];
  __shared__ int  mshare;

  const int tid  = threadIdx.x;
  const int b    = blockIdx.x & 63;
  const int mode = blockIdx.x >> 6;  // 0: xy, 1: xx, 2: yy
  const float* yp = y_pred + b * TLEN;
  const float* yt = y_true + b * TLEN;

  __builtin_prefetch(yp + tid, 0, 0);   // global_prefetch_b8
  __builtin_prefetch(yt + tid, 0, 0);

  // ---- lengths: len_p, len_t, m = min ----
  int cp = 0, ctn = 0;
  for (int t = tid; t < TLEN; t += 256) {
    cp  += (yp[t] != PADV) ? 1 : 0;
    ctn += (yt[t] != PADV) ? 1 : 0;
  }
  redi[tid] = cp;
  redi[256 + tid] = ctn;
  __syncthreads();
  for (int s = 128; s > 0; s >>= 1) {
    if (tid < s) { redi[tid] += redi[tid + s]; redi[256 + tid] += redi[256 + tid + s]; }
    __syncthreads();
  }
  if (tid == 0) {
    int lp = redi[0], lt = redi[256];
    mshare = lp < lt ? lp : lt;
  }
  __syncthreads();
  const int m = mshare;

  // ---- build trimmed/homogenized arrays (PAD remapped to SENT; exactly
  //      equivalent in fp32 since pad<->valid couplings underflow to 0) ----
  const float logm = __logf((float)m);   // natural log
  for (int j = tid; j < TLEN; j += 256) {
    float xv = 0.0f, yv = 0.0f, wv = -1.0e9f;
    if (j < WLEN) {
      const bool trim = j < (m - 2);
      const float ytj = yt[j + 1];
      const float ypj = yp[j + 1];
      xv = trim ? ytj : SENT;           // x = y_pred_homo (true vals where trimmed)
      yv = trim ? ypj : SENT;           // y = y_true_homo (pred vals where trimmed)
      wv = (j < m) ? (-logm) : -1.0e9f; // la == lb == log(w)
    }
    xs[j] = xv; ys[j] = yv; wl[j] = wv;
    fa[j] = 0.0f; ga[j] = 0.0f;
  }
  __syncthreads();

  // ---- timing partial sums (mode-0 blocks only) ----
  if (mode == 0) {
    float tacc = 0.0f;
    for (int j = tid; j < WLEN; j += 256) {
      if (j < m - 2) { float d = yp[j + 1] - yt[j + 1]; tacc += d * d; }
    }
    red[tid] = tacc;
    __syncthreads();
    for (int s = 128; s > 0; s >>= 1) {
      if (tid < s) red[tid] += red[tid + s];
      __syncthreads();
    }
    if (tid == 0) {
      ws[192 + b] = red[0];
      int nv = m - 2; if (nv > WLEN) nv = WLEN;
      ws[256 + b] = (float)nv;
    }
    __syncthreads();
  }

  // ---- Sinkhorn iterations ----
  if (mode == 0) {
    for (int it = 0; it < NITER; ++it) {
      softmin_pass(xs, ys, wl, ga, fa, qv, tid);  // f = softmin(C_xy, lb, g)
      softmin_pass(ys, xs, wl, fa, ga, qv, tid);  // g = softmin(C_yx, la, f)
    }
  } else {
    const float* arr = (mode == 1) ? xs : ys;
    for (int it = 0; it < NITER; ++it) {
      softmin_pass(arr, arr, wl, fa, ga, qv, tid); // s = softmin(C, wlog, p)
      for (int j = tid; j < TLEN; j += 256) fa[j] = 0.5f * (fa[j] + ga[j]);
      __syncthreads();
    }
  }

  // ---- per-batch OT partial ----
  float acc = 0.0f;
  int mv = m < WLEN ? m : WLEN;
  const float invm = 1.0f / (float)m;
  if (mode == 0) {
    for (int i = tid; i < TLEN; i += 256)
      if (i < mv) acc += invm * (fa[i] + ga[i]);          // sum(a*f)+sum(b*g)
  } else {
    for (int i = tid; i < TLEN; i += 256)
      if (i < mv) acc += 2.0f * invm * fa[i];             // 2*sum(w*p)
  }
  red[tid] = acc;
  __syncthreads();
  for (int s = 128; s > 0; s >>= 1) {
    if (tid < s) red[tid] += red[tid + s];
    __syncthreads();
  }
  if (tid == 0) ws[mode * 64 + b] = red[0];
}

__global__ __launch_bounds__(64) void finalize_kernel(
    const float* __restrict__ lp, const float* __restrict__ lt,
    const float* __restrict__ ws, float* __restrict__ out) {
  __shared__ float sd[64], sl[64], st[64], sn[64];
  const int t = threadIdx.x;
  const float dv = ws[t] - 0.5f * (ws[64 + t] + ws[128 + t]);
  const float dl = lp[t] - lt[t];
  sd[t] = dv; sl[t] = dl * dl; st[t] = ws[192 + t]; sn[t] = ws[256 + t];
  __syncthreads();
  for (int s = 32; s > 0; s >>= 1) {
    if (t < s) { sd[t] += sd[t+s]; sl[t] += sl[t+s]; st[t] += st[t+s]; sn[t] += sn[t+s]; }
    __syncthreads();
  }
  if (t == 0) {
    const float length_loss = sl[0] * (1.0f / 64.0f);
    const float timing_loss = st[0] / sn[0];
    const float distrib_loss = sd[0] * (1.0f / 64.0f);
    out[0] = timing_loss + length_loss + distrib_loss;
    out[1] = length_loss;
    out[2] = timing_loss;
  }
}

extern "C" void kernel_launch(void* const* d_in, const int* in_sizes, int n_in,
                              void* d_out, int out_size, void* d_ws, size_t ws_size,
                              hipStream_t stream) {
  const float* y_pred = (const float*)d_in[0];
  const float* y_true = (const float*)d_in[1];
  const float* len_p  = (const float*)d_in[2];
  const float* len_t  = (const float*)d_in[3];
  float* ws  = (float*)d_ws;
  float* out = (float*)d_out;
  // 192 blocks = 64 batches x {xy, xx, yy}; 256 threads = 8 wave32
  sinkhorn_kernel<<<192, 256, 0, stream>>>(y_pred, y_true, ws);
  finalize_kernel<<<1, 64, 0, stream>>>(len_p, len_t, ws, out);
}